// _ProposalLayer_34093450395664
// MI455X (gfx1250) — compile-verified
//
#include <hip/hip_runtime.h>
#include <stdint.h>

#define NMS_T 0.7f

typedef __attribute__((ext_vector_type(2))) float v2f;
typedef __attribute__((ext_vector_type(8))) float v8f;

__device__ __forceinline__ float diou3d(const float b1[6], const float b2[6]) {
  // box layout: x1,y1,x2,y2,z1,z2
  float w1 = b1[2]-b1[0], h1 = b1[3]-b1[1], d1 = b1[5]-b1[4];
  float w2 = b2[2]-b2[0], h2 = b2[3]-b2[1], d2 = b2[5]-b2[4];
  float area1 = w1*h1*d1, area2 = w2*h2*d2;
  float cx1 = (b1[2]+b1[0])*0.5f, cy1 = (b1[3]+b1[1])*0.5f, cz1 = (b1[4]+b1[5])*0.5f;
  float cx2 = (b2[2]+b2[0])*0.5f, cy2 = (b2[3]+b2[1])*0.5f, cz2 = (b2[4]+b2[5])*0.5f;
  float iw = fmaxf(fminf(b1[2],b2[2]) - fmaxf(b1[0],b2[0]), 0.0f);
  float ih = fmaxf(fminf(b1[3],b2[3]) - fmaxf(b1[1],b2[1]), 0.0f);
  float id = fmaxf(fminf(b1[5],b2[5]) - fmaxf(b1[4],b2[4]), 0.0f);
  float inter = iw*ih*id;
  float dx = cx2-cx1, dy = cy2-cy1, dz = cz2-cz1;
  float inter_diag = dx*dx + dy*dy + dz*dz;
  float ow = fmaxf(fmaxf(b1[2],b2[2]) - fminf(b1[0],b2[0]), 0.0f);
  float oh = fmaxf(fmaxf(b1[3],b2[3]) - fminf(b1[1],b2[1]), 0.0f);
  float od = fmaxf(fmaxf(b1[5],b2[5]) - fminf(b1[4],b2[4]), 0.0f);
  float outer_diag = ow*ow + oh*oh + od*od + 1e-7f;
  float uni = area1 + area2 - inter + 1e-7f;
  float v = inter/uni - inter_diag/outer_diag;
  return fminf(fmaxf(v, -1.0f), 1.0f);
}

// Kernel 1: stable descending rank sort (matches jnp.argsort(-scores), stable).
__global__ void k_ranksort(const float* __restrict__ boxes,
                           const float* __restrict__ scores,
                           float* __restrict__ sboxes,
                           float* __restrict__ sscores,
                           float* __restrict__ out_scores, int N) {
  __shared__ float chunk[1024];
  int i = blockIdx.x * blockDim.x + threadIdx.x;
  float si = (i < N) ? scores[i] : 0.0f;
  int rank = 0;
  for (int base = 0; base < N; base += 1024) {
    int cnt = min(1024, N - base);
    __syncthreads();
    for (int t = threadIdx.x; t < cnt; t += blockDim.x) chunk[t] = scores[base + t];
    __syncthreads();
    if (i < N) {
      for (int k = 0; k < cnt; ++k) {
        float sj = chunk[k];
        int j = base + k;
        rank += (sj > si) || (sj == si && j < i);
      }
    }
  }
  if (i < N) {
    sscores[rank] = si;
    out_scores[rank] = si;
#pragma unroll
    for (int c = 0; c < 6; ++c) sboxes[rank * 6 + c] = boxes[i * 6 + c];
  }
}

// Kernel 2: suppression bitmask. mask[i][w] bit b set iff j=w*32+b, j>i, diou(i,j) >= 0.7
__global__ void k_mask(const float* __restrict__ sboxes,
                       uint32_t* __restrict__ mask, int N, int W) {
  int i = blockIdx.x;
  int lane = threadIdx.x & 31;
  int wv = threadIdx.x >> 5;
  int nwaves = blockDim.x >> 5;
  float bi[6];
#pragma unroll
  for (int c = 0; c < 6; ++c) bi[c] = sboxes[i * 6 + c];
  for (int w = wv; w < W; w += nwaves) {
    uint32_t word = 0;
    if (w * 32 + 31 > i) {  // skip words entirely in the lower triangle
      int j = w * 32 + lane;
      bool pred = false;
      if (j < N && j > i) {
        float bj[6];
#pragma unroll
        for (int c = 0; c < 6; ++c) bj[c] = sboxes[j * 6 + c];
        pred = diou3d(bi, bj) >= NMS_T;
      }
      word = (uint32_t)__ballot(pred);
    }
    if (lane == 0) mask[(size_t)i * W + w] = word;
  }
}

// Kernel 3: chunked serial greedy-NMS reduce (fixed point of cluster-NMS).
__global__ void k_nms_serial(const uint32_t* __restrict__ mask,
                             uint32_t* __restrict__ keep, int N, int W) {
  __shared__ uint32_t remv[256];  // W <= 256 (N <= 8192)
  __shared__ uint32_t tmp[32];
  __shared__ uint32_t kws;
  int t = threadIdx.x;
  for (int w = t; w < W; w += blockDim.x) remv[w] = 0u;
  __syncthreads();
  for (int c = 0; c < W; ++c) {
    if (t < 32) {
      int i = c * 32 + t;
      tmp[t] = (i < N) ? mask[(size_t)i * W + c] : 0u;
    }
    __syncthreads();
    if (t == 0) {
      uint32_t r = remv[c], kw = 0u;
      for (int k = 0; k < 32; ++k) {
        int i = c * 32 + k;
        if (i < N && !((r >> k) & 1u)) { kw |= (1u << k); r |= tmp[k]; }
      }
      remv[c] = r;
      kws = kw;
    }
    __syncthreads();
    uint32_t kw = kws;
    if (t < 32) {
      int i = c * 32 + t;
      if (i < N) keep[i] = (kw >> t) & 1u;
    }
    if (t < W) {
      uint32_t acc = 0u;
      for (int k = 0; k < 32; ++k)
        if (kw & (1u << k)) acc |= mask[(size_t)(c * 32 + k) * W + t];
      remv[t] |= acc;
    }
    __syncthreads();
  }
}

// Kernel 4: merged boxes via V_WMMA_F32_16X16X4_F32.
// D (16x16 f32, cols 0..5 = weighted box sums, col 6 = weight sum) accumulated over K=4 j-slices.
// A layout (16x4 f32): VGPR v, lane L -> (m = L%16, k = v + 2*(L>=16))
// B layout (4x16 f32): VGPR v, lane L -> (k = v + 2*(L>=16), n = L%16)
// D layout          : VGPR r, lane L -> (m = r + 8*(L>=16), n = L%16)
__global__ void k_merge(const float* __restrict__ sboxes,
                        const float* __restrict__ sscores,
                        const uint32_t* __restrict__ keep,
                        float* __restrict__ out_boxes,
                        float* __restrict__ out_keep, int N) {
  __shared__ float red[8][16][8];
  __shared__ float tot[16][8];
  const int i0 = blockIdx.x * 16;
  const int tid = threadIdx.x;
  const int lane = tid & 31;
  const int wv = tid >> 5;      // 8 waves per block
  const int m = lane & 15;
  const int hf = lane >> 4;     // 0: K=0,1 ; 1: K=2,3
  const int i = i0 + m;
  const bool valid_i = (i < N);
  float bi[6];
#pragma unroll
  for (int c = 0; c < 6; ++c) bi[c] = valid_i ? sboxes[i * 6 + c] : 0.0f;
  const bool kp = valid_i && (keep[i] != 0u);

  v8f acc = {0.f, 0.f, 0.f, 0.f, 0.f, 0.f, 0.f, 0.f};

  const int jbase = i0 & ~31;   // j-blocks fully below the diagonal are skipped
  for (int j0 = jbase + wv * 4; j0 < N; j0 += 32) {
    int ja = j0 + hf * 2;       // this lane's two K-slots (A and B share them)
    int jb = ja + 1;
    float wA = 0.f, wB = 0.f, b0v = 0.f, b1v = 0.f;
    if (ja < N) {
      float bj[6];
#pragma unroll
      for (int c = 0; c < 6; ++c) bj[c] = sboxes[ja * 6 + c];
      if (kp && ja > i) {
        float d = diou3d(bi, bj);
        if (d > NMS_T) wA = sscores[ja] * d;
      }
      b0v = (m < 6) ? sboxes[ja * 6 + m] : ((m == 6) ? 1.0f : 0.0f);
    }
    if (jb < N) {
      float bj[6];
#pragma unroll
      for (int c = 0; c < 6; ++c) bj[c] = sboxes[jb * 6 + c];
      if (kp && jb > i) {
        float d = diou3d(bi, bj);
        if (d > NMS_T) wB = sscores[jb] * d;
      }
      b1v = (m < 6) ? sboxes[jb * 6 + m] : ((m == 6) ? 1.0f : 0.0f);
    }
    v2f a; a.x = wA; a.y = wB;
    v2f b; b.x = b0v; b.y = b1v;
    acc = __builtin_amdgcn_wmma_f32_16x16x4_f32(
        false, a, false, b, (short)0, acc, false, false);
  }

  // cross-wave reduction of the 16x7 tile
  if (m < 7) {
#pragma unroll
    for (int r = 0; r < 8; ++r) red[wv][r + 8 * hf][m] = acc[r];
  }
  __syncthreads();
  if (tid < 112) {
    int mm = tid / 7, nn = tid % 7;
    float s = 0.f;
#pragma unroll
    for (int w = 0; w < 8; ++w) s += red[w][mm][nn];
    tot[mm][nn] = s;
  }
  __syncthreads();
  if (tid < 16) {
    int i2 = i0 + tid;
    if (i2 < N) {
      float si = sscores[i2];
      float denom = tot[tid][6] + si;  // weight sum incl. diagonal s_i
#pragma unroll
      for (int c = 0; c < 6; ++c)
        out_boxes[i2 * 6 + c] = (tot[tid][c] + si * sboxes[i2 * 6 + c]) / denom;
      out_keep[i2] = (keep[i2] != 0u) ? 1.0f : 0.0f;
    }
  }
}

extern "C" void kernel_launch(void* const* d_in, const int* in_sizes, int n_in,
                              void* d_out, int out_size, void* d_ws, size_t ws_size,
                              hipStream_t stream) {
  const float* boxes  = (const float*)d_in[0];
  const float* scores = (const float*)d_in[1];
  const int N = in_sizes[0] / 6;
  const int W = (N + 31) / 32;

  auto align256 = [](size_t x) { return (x + 255) & ~(size_t)255; };
  char* ws = (char*)d_ws;
  size_t off = 0;
  float* sboxes = (float*)(ws + off);   off += align256((size_t)N * 6 * sizeof(float));
  float* sscores = (float*)(ws + off);  off += align256((size_t)N * sizeof(float));
  uint32_t* keep = (uint32_t*)(ws + off); off += align256((size_t)N * sizeof(uint32_t));
  uint32_t* mask = (uint32_t*)(ws + off); off += align256((size_t)N * W * sizeof(uint32_t));
  (void)ws_size; (void)n_in; (void)out_size;

  float* out_boxes  = (float*)d_out;          // [N,6]
  float* out_keep   = out_boxes + (size_t)N * 6;  // [N]
  float* out_scores = out_keep + N;           // [N,1]

  k_ranksort<<<(N + 255) / 256, 256, 0, stream>>>(boxes, scores, sboxes, sscores,
                                                  out_scores, N);
  k_mask<<<N, 256, 0, stream>>>(sboxes, mask, N, W);
  k_nms_serial<<<1, 256, 0, stream>>>(mask, keep, N, W);
  k_merge<<<(N + 15) / 16, 256, 0, stream>>>(sboxes, sscores, keep, out_boxes,
                                             out_keep, N);
}